// VGAE_MLP_22986664968329
// MI455X (gfx1250) — compile-verified
//
#include <hip/hip_runtime.h>
#include <math.h>

#define NN    50000
#define EE    800000
#define INDIM 32
#define HID   64
#define ZD    32

typedef __attribute__((ext_vector_type(2))) float v2f;
typedef __attribute__((ext_vector_type(8))) float v8f;

__device__ __forceinline__ void atomAddF(float* p, float v) {
  (void)__hip_atomic_fetch_add(p, v, __ATOMIC_RELAXED, __HIP_MEMORY_SCOPE_AGENT);
}

// ---------------- degree / symmetric norm ----------------
__global__ void k_init_deg(float* __restrict__ deg) {
  int i = blockIdx.x * blockDim.x + threadIdx.x;
  if (i < NN) deg[i] = 1.0f;  // self-loop contribution
}
__global__ void k_deg_edges(const int* __restrict__ dst, float* __restrict__ deg) {
  int e = blockIdx.x * blockDim.x + threadIdx.x;
  if (e < EE) atomAddF(&deg[dst[e]], 1.0f);
}
__global__ void k_dinv(float* __restrict__ deg) {
  int i = blockIdx.x * blockDim.x + threadIdx.x;
  if (i < NN) deg[i] = rsqrtf(fmaxf(deg[i], 1.0f));
}

// ---------------- fp32 WMMA GEMM: out[M,N] = A[M,K] @ W[K,N] ----------------
// One wave per 16x16 output tile; V_WMMA_F32_16X16X4_F32, K stepped by 4.
// A layout (ISA 7.12.2): lanes 0-15 hold M=lane, K={0,1} in v0/v1; lanes 16-31 K={2,3}.
// B mirrored (N=lane&15, K split across lane halves). C/D: 8 VGPRs, M split by half-wave.
__global__ void k_gemm_wmma(const float* __restrict__ A, const float* __restrict__ W,
                            float* __restrict__ out, int K, int N) {
  const int tileM = blockIdx.x;        // M/16 tiles (50000/16 = 3125 exact)
  const int tileN = threadIdx.y;       // one wave per N-tile
  const int lane  = threadIdx.x;       // wave32
  const int half  = lane >> 4;
  const int l16   = lane & 15;
  const int klo   = half << 1;         // 0 or 2

  v8f c = {};
  const float* arow = A + (long)(tileM * 16 + l16) * K;
  const float* wcol = W + tileN * 16 + l16;
  for (int k = 0; k < K; k += 4) {
    v2f a, b;
    a.x = arow[k + klo + 0];
    a.y = arow[k + klo + 1];
    b.x = wcol[(long)(k + klo + 0) * N];
    b.y = wcol[(long)(k + klo + 1) * N];
    c = __builtin_amdgcn_wmma_f32_16x16x4_f32(false, a, false, b, (short)0, c,
                                              false, false);
  }
  float* op = out + (long)(tileM * 16 + half * 8) * N + tileN * 16 + l16;
#pragma unroll
  for (int r = 0; r < 8; ++r) op[(long)r * N] = c[r];
}

// ---------------- utility ----------------
__global__ void k_zero(float* __restrict__ p, long n) {
  long i = (long)blockIdx.x * blockDim.x + threadIdx.x;
  if (i < n) p[i] = 0.0f;
}

// ---------------- edge scatter: acc[dst] += h[src] * dinv[src]*dinv[dst] ----------------
// 16 lanes per edge, each lane moves C/16 contiguous floats (128b/64b gather).
template <int C>
__global__ void k_scatter(const float* __restrict__ h, const int* __restrict__ src,
                          const int* __restrict__ dst, const float* __restrict__ dinv,
                          float* __restrict__ acc) {
  long tid = (long)blockIdx.x * blockDim.x + threadIdx.x;
  long e   = tid >> 4;
  int  sub = (int)(tid & 15);
  if (e >= EE) return;
  int s = src[e], d = dst[e];
  float w = dinv[s] * dinv[d];
  constexpr int V = C / 16;
  const float* hp = h + (long)s * C + sub * V;
  float* ap = acc + (long)d * C + sub * V;
  if constexpr (V == 4) {
    float4 v = *(const float4*)hp;
    atomAddF(ap + 0, v.x * w);
    atomAddF(ap + 1, v.y * w);
    atomAddF(ap + 2, v.z * w);
    atomAddF(ap + 3, v.w * w);
  } else {
    float2 v = *(const float2*)hp;
    atomAddF(ap + 0, v.x * w);
    atomAddF(ap + 1, v.y * w);
  }
}

// ---------------- epilogue: out = act(acc + lin*dinv^2 (self loop) + bias) ----------------
template <bool RELU, int C>
__global__ void k_epilogue(const float* __restrict__ acc, const float* __restrict__ lin,
                           const float* __restrict__ dinv, const float* __restrict__ bias,
                           float* __restrict__ out) {
  long i = (long)blockIdx.x * blockDim.x + threadIdx.x;
  if (i >= (long)NN * C) return;
  int node = (int)(i / C);
  int ch   = (int)(i % C);
  float di = dinv[node];
  float v  = acc[i] + lin[i] * di * di + bias[ch];
  out[i]   = RELU ? fmaxf(v, 0.0f) : v;
}

// ---------------- reparameterize: z = mu + exp(0.5*logvar)*eps ----------------
__global__ void k_reparam(const float* __restrict__ mu, const float* __restrict__ lv,
                          const float* __restrict__ eps, float* __restrict__ z) {
  long i = (long)blockIdx.x * blockDim.x + threadIdx.x;
  if (i >= (long)NN * ZD) return;
  z[i] = mu[i] + expf(0.5f * lv[i]) * eps[i];
}

extern "C" void kernel_launch(void* const* d_in, const int* in_sizes, int n_in,
                              void* d_out, int out_size, void* d_ws, size_t ws_size,
                              hipStream_t stream) {
  const float* x   = (const float*)d_in[0];
  const int*   ei  = (const int*)d_in[1];
  const float* W1  = (const float*)d_in[2];
  const float* b1  = (const float*)d_in[3];
  const float* W2  = (const float*)d_in[4];
  const float* b2  = (const float*)d_in[5];
  const float* Wmu = (const float*)d_in[6];
  const float* bmu = (const float*)d_in[7];
  const float* Wlv = (const float*)d_in[8];
  const float* blv = (const float*)d_in[9];
  const float* eps = (const float*)d_in[10];
  const int* src = ei;
  const int* dst = ei + EE;

  float* ws   = (float*)d_ws;
  float* dinv = ws;                          // NN
  float* lin  = dinv + NN;                   // NN*HID (gemm result, reused)
  float* acc  = lin + (long)NN * HID;        // NN*HID (scatter accumulator)
  float* h1   = acc + (long)NN * HID;        // NN*HID
  float* h2   = h1 + (long)NN * HID;         // NN*HID

  float* zout = (float*)d_out;               // NN*ZD
  float* muo  = zout + (long)NN * ZD;        // NN*ZD
  float* lvo  = muo + (long)NN * ZD;         // NN*ZD

  const int B = 256;
  const int gN  = (NN + B - 1) / B;
  const int gE  = (EE + B - 1) / B;
  const int gE16 = (int)(((long)EE * 16 + B - 1) / B);
  const int gH  = (int)(((long)NN * HID + B - 1) / B);
  const int gZ  = (int)(((long)NN * ZD + B - 1) / B);

  // symmetric normalization (shared by all 4 convs)
  k_init_deg<<<gN, B, 0, stream>>>(dinv);
  k_deg_edges<<<gE, B, 0, stream>>>(dst, dinv);
  k_dinv<<<gN, B, 0, stream>>>(dinv);

  // conv1: relu(gcn(x, W1, b1)) -> h1
  k_gemm_wmma<<<dim3(NN / 16), dim3(32, HID / 16), 0, stream>>>(x, W1, lin, INDIM, HID);
  k_zero<<<gH, B, 0, stream>>>(acc, (long)NN * HID);
  k_scatter<HID><<<gE16, B, 0, stream>>>(lin, src, dst, dinv, acc);
  k_epilogue<true, HID><<<gH, B, 0, stream>>>(acc, lin, dinv, b1, h1);

  // conv2: relu(gcn(h1, W2, b2)) -> h2
  k_gemm_wmma<<<dim3(NN / 16), dim3(32, HID / 16), 0, stream>>>(h1, W2, lin, HID, HID);
  k_zero<<<gH, B, 0, stream>>>(acc, (long)NN * HID);
  k_scatter<HID><<<gE16, B, 0, stream>>>(lin, src, dst, dinv, acc);
  k_epilogue<true, HID><<<gH, B, 0, stream>>>(acc, lin, dinv, b2, h2);

  // conv_mu: gcn(h2, W_mu, b_mu) -> muo (d_out slot 1)
  k_gemm_wmma<<<dim3(NN / 16), dim3(32, ZD / 16), 0, stream>>>(h2, Wmu, lin, HID, ZD);
  k_zero<<<gZ, B, 0, stream>>>(acc, (long)NN * ZD);
  k_scatter<ZD><<<gE16, B, 0, stream>>>(lin, src, dst, dinv, acc);
  k_epilogue<false, ZD><<<gZ, B, 0, stream>>>(acc, lin, dinv, bmu, muo);

  // conv_lv: gcn(h2, W_lv, b_lv) -> lvo (d_out slot 2)
  k_gemm_wmma<<<dim3(NN / 16), dim3(32, ZD / 16), 0, stream>>>(h2, Wlv, lin, HID, ZD);
  k_zero<<<gZ, B, 0, stream>>>(acc, (long)NN * ZD);
  k_scatter<ZD><<<gE16, B, 0, stream>>>(lin, src, dst, dinv, acc);
  k_epilogue<false, ZD><<<gZ, B, 0, stream>>>(acc, lin, dinv, blv, lvo);

  // z = mu + exp(0.5*logvar)*eps -> zout (d_out slot 0)
  k_reparam<<<gZ, B, 0, stream>>>(muo, lvo, eps, zout);
}